// DisableNeighborTOFs_25494925869704
// MI455X (gfx1250) — compile-verified
//
#include <hip/hip_runtime.h>

// Pure streaming masked copy: out[r, c] = disabled(c) ? 0 : img[r, c]
// disabled(c) = ((c - start) mod 16) < (2 + count_offset)
//
// Bandwidth-bound (1 GiB traffic, ~46 us floor at 23.3 TB/s). WMMA/TDM are
// deliberately NOT used: they would force narrower, permuted accesses on a
// kernel whose only cost is HBM traffic. We use wave32 b128 non-temporal
// loads/stores (streams exceed the 192 MB L2).

typedef float v4f __attribute__((ext_vector_type(4)));

#define TOFS 16

__global__ __launch_bounds__(256) void disable_neighbor_tofs_kernel(
    const float* __restrict__ img,
    const int*   __restrict__ count_offset,
    const int*   __restrict__ start,
    float*       __restrict__ out,
    int rows)
{
    const int row = blockIdx.x * (int)blockDim.x + (int)threadIdx.x;
    if (row >= rows) return;

    // Uniform scalars (compiler scalarizes: s_load + SALU).
    const int count = 2 + count_offset[0];   // in [2, 5]
    const int s     = start[0];              // in [0, 16)

    const v4f* __restrict__ src = (const v4f*)(img + (size_t)row * TOFS);
    v4f*       __restrict__ dst = (v4f*)(out + (size_t)row * TOFS);

    // 4 x global_load_b128, non-temporal (no reuse; input > L2).
    v4f v[4];
    v[0] = __builtin_nontemporal_load(src + 0);
    v[1] = __builtin_nontemporal_load(src + 1);
    v[2] = __builtin_nontemporal_load(src + 2);
    v[3] = __builtin_nontemporal_load(src + 3);

    // 16 selects with wave-uniform conditions -> SALU cmp + v_cndmask.
#pragma unroll
    for (int g = 0; g < 4; ++g) {
#pragma unroll
        for (int k = 0; k < 4; ++k) {
            const int c = g * 4 + k;
            // two's-complement AND gives correct mod-16 for negative c - s
            const bool disabled = (((c - s) & (TOFS - 1)) < count);
            v[g][k] = disabled ? 0.0f : v[g][k];
        }
    }

    // 4 x global_store_b128, non-temporal (pure write stream).
    __builtin_nontemporal_store(v[0], dst + 0);
    __builtin_nontemporal_store(v[1], dst + 1);
    __builtin_nontemporal_store(v[2], dst + 2);
    __builtin_nontemporal_store(v[3], dst + 3);
}

extern "C" void kernel_launch(void* const* d_in, const int* in_sizes, int n_in,
                              void* d_out, int out_size, void* d_ws, size_t ws_size,
                              hipStream_t stream)
{
    const float* img          = (const float*)d_in[0];
    const int*   count_offset = (const int*)  d_in[1];
    const int*   start        = (const int*)  d_in[2];
    float*       out          = (float*)      d_out;

    const int rows  = in_sizes[0] / TOFS;          // 8,388,608
    const int block = 256;                          // 8 wave32 waves / block
    const int grid  = (rows + block - 1) / block;   // 32,768 blocks

    disable_neighbor_tofs_kernel<<<grid, block, 0, stream>>>(
        img, count_offset, start, out, rows);
}